// Model_65231963291698
// MI455X (gfx1250) — compile-verified
//
#include <hip/hip_runtime.h>

// ---------------------------------------------------------------------------
// Hetero-GraphSAGE (2-layer) + dot predictor for MI455X (gfx1250, wave32).
// Dense GEMMs: V_WMMA_F32_16X16X4_F32 with LDS-staged A/X tiles (coalesced
// global fills + conflict-free ds_load fragment reads). Segment-mean scatters
// use f32 global atomics. Memory-bound workload -> f32 kept throughout.
// ---------------------------------------------------------------------------

typedef __attribute__((ext_vector_type(2))) float v2f;
typedef __attribute__((ext_vector_type(8))) float v8f;

static __device__ __forceinline__ void atomic_add_f32(float* p, float v) {
    __hip_atomic_fetch_add(p, v, __ATOMIC_RELAXED, __HIP_MEMORY_SCOPE_AGENT);
}

// ---------------------------------------------------------------------------
// Zero-fill (float4 granularity; all region sizes are multiples of 4 floats)
// ---------------------------------------------------------------------------
__global__ void fill0(float4* __restrict__ p, long n4) {
    long t = (long)blockIdx.x * blockDim.x + threadIdx.x;
    if (t < n4) p[t] = make_float4(0.f, 0.f, 0.f, 0.f);
}

// ---------------------------------------------------------------------------
// xu[i,:] = user_emb_w[user_node_id[i],:]   (D = 64, float4 per thread)
// ---------------------------------------------------------------------------
__global__ void gather_rows(const float* __restrict__ emb,
                            const int* __restrict__ idx,
                            float* __restrict__ out, long nthreads) {
    long t = (long)blockIdx.x * blockDim.x + threadIdx.x;
    if (t >= nthreads) return;
    long row = t >> 4;
    int  j   = (int)(t & 15) << 2;
    long s   = idx[row];
    *(float4*)(out + row * 64 + j) = *(const float4*)(emb + s * 64 + j);
}

// ---------------------------------------------------------------------------
// Fused SAGE GEMM:
//   OUT[N,64] = act( (SCALE ? A/max(cnt,1) : A) @ WL  [+ X @ WR]  + bias )
//
// Block = 128 threads (4 waves); block -> 16 rows x 64 cols, wave -> one
// 16x16 tile. A (and X) 16-row tiles are staged in LDS with coalesced float4
// global loads, then WMMA fragments are read from LDS:
//   A-fragment (16x4 f32): lane l holds A[l%16][k + 2*(l/16) + {0,1}]  (v2f)
//   B-fragment (4x16 f32): lane l holds W[k + 2*(l/16) + {0,1}][col0 + l%16]
//   C/D (16x16 f32): VGPR r of lane l = OUT[row0 + r + 8*(l/16)][col0 + l%16]
// LDS row stride S chosen with S/4 odd -> banks (4m + c) mod 64 distinct for
// the 16 rows of a half-wave; lanes 16-31 are offset by 2 banks (disjoint).
// ---------------------------------------------------------------------------
template <int K1, int S1, bool SCALE, bool HAS2, bool RELU>
__global__ __launch_bounds__(128)
void sage_gemm(const float* __restrict__ A, int lda,
               const float* __restrict__ WL,      // [K1, 64]
               const float* __restrict__ cnt,     // [N]  (SCALE)
               const float* __restrict__ X,       // [N, 64] (HAS2)
               const float* __restrict__ WR,      // [64, 64] (HAS2)
               const float* __restrict__ bias,    // [64]
               float* __restrict__ OUT)           // [N, 64]
{
    __shared__ float As[16 * S1];
    __shared__ float Xs[HAS2 ? 16 * 68 : 4];

    const int tid  = threadIdx.x;
    const int wave = tid >> 5;
    const int lane = tid & 31;
    const int m    = lane & 15;            // row (A) / col (B) within tile
    const int kh   = (lane >> 4) << 1;     // 0 or 2 : K sub-offset
    const long row0 = (long)blockIdx.x * 16;
    const int  col0 = wave << 4;

    // --- stage A tile (16 x K1) into LDS, coalesced float4 ---
    constexpr int C4A = K1 / 4;            // float4 chunks per row
    for (int idx = tid; idx < 16 * C4A; idx += 128) {
        int r = idx / C4A, c = (idx % C4A) << 2;
        float4 v = *(const float4*)(A + (row0 + r) * (long)lda + c);
        *(float4*)&As[r * S1 + c] = v;
    }
    // --- stage X tile (16 x 64) into LDS ---
    if (HAS2) {
        for (int idx = tid; idx < 16 * 16; idx += 128) {
            int r = idx >> 4, c = (idx & 15) << 2;
            float4 v = *(const float4*)(X + (row0 + r) * 64 + c);
            *(float4*)&Xs[r * 68 + c] = v;
        }
    }
    __syncthreads();

    v8f acc = {};
    float rcp = 1.0f;
    if (SCALE) rcp = 1.0f / fmaxf(cnt[row0 + m], 1.0f);

#pragma unroll
    for (int k = 0; k < K1; k += 4) {
        v2f a = *(const v2f*)(&As[m * S1 + k + kh]);
        if (SCALE) { a.x *= rcp; a.y *= rcp; }
        v2f b;
        b.x = WL[(k + kh)     * 64 + col0 + m];
        b.y = WL[(k + kh + 1) * 64 + col0 + m];
        acc = __builtin_amdgcn_wmma_f32_16x16x4_f32(
                  false, a, false, b, (short)0, acc, false, false);
    }
    if (HAS2) {
#pragma unroll
        for (int k = 0; k < 64; k += 4) {
            v2f a = *(const v2f*)(&Xs[m * 68 + k + kh]);
            v2f b;
            b.x = WR[(k + kh)     * 64 + col0 + m];
            b.y = WR[(k + kh + 1) * 64 + col0 + m];
            acc = __builtin_amdgcn_wmma_f32_16x16x4_f32(
                      false, a, false, b, (short)0, acc, false, false);
        }
    }
    const float bv = bias[col0 + m];
#pragma unroll
    for (int r = 0; r < 8; ++r) {
        long row = row0 + r + ((lane >> 4) << 3);
        float v = acc[r] + bv;
        if (RELU) v = fmaxf(v, 0.0f);
        OUT[row * 64 + col0 + m] = v;
    }
}

// ---------------------------------------------------------------------------
// Segment-sum counts: cnt_u += 1 at edge_src, cnt_i += 1 at edge_dst
// ---------------------------------------------------------------------------
__global__ void edge_counts(const int* __restrict__ src,
                            const int* __restrict__ dst,
                            float* __restrict__ cu, float* __restrict__ ci,
                            long E) {
    long e = (long)blockIdx.x * blockDim.x + threadIdx.x;
    if (e >= E) return;
    atomic_add_f32(&cu[src[e]], 1.0f);
    atomic_add_f32(&ci[dst[e]], 1.0f);
}

// ---------------------------------------------------------------------------
// out[didx[e], :] += feat[sidx[e], :]   (D = 64; 16 threads/edge, float4 each)
// ---------------------------------------------------------------------------
__global__ void scatter_add64(const float* __restrict__ feat,
                              const int* __restrict__ sidx,
                              const int* __restrict__ didx,
                              float* __restrict__ out, long E) {
    long t = (long)blockIdx.x * blockDim.x + threadIdx.x;
    long e = t >> 4;
    if (e >= E) return;
    int  j = (int)(t & 15) << 2;
    long s = sidx[e], d = didx[e];
    float4 v = *(const float4*)(feat + s * 64 + j);
    float* o = out + d * 64 + j;
    atomic_add_f32(o + 0, v.x);
    atomic_add_f32(o + 1, v.y);
    atomic_add_f32(o + 2, v.z);
    atomic_add_f32(o + 3, v.w);
}

// ---------------------------------------------------------------------------
// out[e] = dot(hu[ls[e],:], hi[ld[e],:])  -- one wave32 per edge
// ---------------------------------------------------------------------------
__global__ __launch_bounds__(256)
void dot_predict(const float* __restrict__ hu, const float* __restrict__ hi,
                 const int* __restrict__ ls, const int* __restrict__ ld,
                 float* __restrict__ out, long EL) {
    const int wave = threadIdx.x >> 5;
    const int lane = threadIdx.x & 31;
    long e = (long)blockIdx.x * 8 + wave;
    if (e >= EL) return;
    long a = (long)ls[e] * 64;
    long b = (long)ld[e] * 64;
    v2f x = *(const v2f*)(hu + a + lane * 2);
    v2f y = *(const v2f*)(hi + b + lane * 2);
    float p = x.x * y.x + x.y * y.y;
#pragma unroll
    for (int off = 16; off; off >>= 1) p += __shfl_xor(p, off);
    if (lane == 0) out[e] = p;
}

// ---------------------------------------------------------------------------
// Host-side orchestration (graph-capture safe: kernel launches only)
// ---------------------------------------------------------------------------
extern "C" void kernel_launch(void* const* d_in, const int* in_sizes, int n_in,
                              void* d_out, int out_size, void* d_ws, size_t ws_size,
                              hipStream_t stream) {
    (void)in_sizes; (void)n_in; (void)out_size; (void)ws_size;
    constexpr long NU = 500000, NI = 100000, F = 300, D = 64;
    constexpr long E = 2000000, EL = 1000000;

    const float* user_emb_w = (const float*)d_in[0];
    const float* item_x     = (const float*)d_in[1];
    const float* item_lin_w = (const float*)d_in[2];
    const float* item_lin_b = (const float*)d_in[3];
    const float* Wl1_ui = (const float*)d_in[4];
    const float* Wr1_ui = (const float*)d_in[5];
    const float* b1_ui  = (const float*)d_in[6];
    const float* Wl1_iu = (const float*)d_in[7];
    const float* Wr1_iu = (const float*)d_in[8];
    const float* b1_iu  = (const float*)d_in[9];
    const float* Wl2_ui = (const float*)d_in[10];
    const float* Wr2_ui = (const float*)d_in[11];
    const float* b2_ui  = (const float*)d_in[12];
    const float* Wl2_iu = (const float*)d_in[13];
    const float* Wr2_iu = (const float*)d_in[14];
    const float* b2_iu  = (const float*)d_in[15];
    const int* user_node_id = (const int*)d_in[16];
    const int* edge_src  = (const int*)d_in[17];
    const int* edge_dst  = (const int*)d_in[18];
    const int* label_src = (const int*)d_in[19];
    const int* label_dst = (const int*)d_in[20];
    float* out = (float*)d_out;

    // Workspace layout (floats). [aggu aggi cntu cnti] contiguous so each
    // zeroing phase is a single fill0 launch.
    float* ws   = (float*)d_ws;
    float* aggu = ws;                   // NU*D
    float* aggi = aggu + NU * D;        // NI*D
    float* cntu = aggi + NI * D;        // NU
    float* cnti = cntu + NU;            // NI
    float* xu   = cnti + NI;            // NU*D
    float* xi   = xu + NU * D;          // NI*D
    float* hu   = xi + NI * D;          // NU*D
    float* hi   = hu + NU * D;          // NI*D
    // Layer-2 outputs reuse xu (hu2) and xi (hi2).

    const int TB = 256;
    const long zero1 = (NU * D + NI * D + NU + NI) / 4;   // aggu+aggi+cntu+cnti
    const long zero2 = (NU * D + NI * D) / 4;             // aggu+aggi

    // 0) zero accumulators + counts
    fill0<<<(int)((zero1 + TB - 1) / TB), TB, 0, stream>>>((float4*)aggu, zero1);
    // 1) xu = user_emb_w[user_node_id]
    gather_rows<<<(int)((NU * 16 + TB - 1) / TB), TB, 0, stream>>>(
        user_emb_w, user_node_id, xu, NU * 16);
    // 2) xi = item_x @ item_lin_w + b        (WMMA, K=300; LDS stride 308: 308/4=77 odd)
    sage_gemm<300, 308, false, false, false><<<(int)(NI / 16), 128, 0, stream>>>(
        item_x, (int)F, item_lin_w, nullptr, nullptr, nullptr, item_lin_b, xi);
    // 3) degree counts (shared by both layers)
    edge_counts<<<(int)((E + TB - 1) / TB), TB, 0, stream>>>(
        edge_src, edge_dst, cntu, cnti, E);
    // 4) layer-1 scatters
    scatter_add64<<<(int)((E * 16 + TB - 1) / TB), TB, 0, stream>>>(
        xu, edge_src, edge_dst, aggi, E);
    scatter_add64<<<(int)((E * 16 + TB - 1) / TB), TB, 0, stream>>>(
        xi, edge_dst, edge_src, aggu, E);
    // 5) layer-1 fused GEMMs: hi = relu(mean@Wl + xi@Wr + b), hu likewise
    sage_gemm<64, 68, true, true, true><<<(int)(NI / 16), 128, 0, stream>>>(
        aggi, (int)D, Wl1_ui, cnti, xi, Wr1_ui, b1_ui, hi);
    sage_gemm<64, 68, true, true, true><<<(int)(NU / 16), 128, 0, stream>>>(
        aggu, (int)D, Wl1_iu, cntu, xu, Wr1_iu, b1_iu, hu);
    // 6) re-zero accumulators, layer-2 scatters
    fill0<<<(int)((zero2 + TB - 1) / TB), TB, 0, stream>>>((float4*)aggu, zero2);
    scatter_add64<<<(int)((E * 16 + TB - 1) / TB), TB, 0, stream>>>(
        hu, edge_src, edge_dst, aggi, E);
    scatter_add64<<<(int)((E * 16 + TB - 1) / TB), TB, 0, stream>>>(
        hi, edge_dst, edge_src, aggu, E);
    // 7) layer-2 fused GEMMs (no relu); outputs reuse xi / xu buffers
    sage_gemm<64, 68, true, true, false><<<(int)(NI / 16), 128, 0, stream>>>(
        aggi, (int)D, Wl2_ui, cnti, hi, Wr2_ui, b2_ui, xi);
    sage_gemm<64, 68, true, true, false><<<(int)(NU / 16), 128, 0, stream>>>(
        aggu, (int)D, Wl2_iu, cntu, hu, Wr2_iu, b2_iu, xu);
    // 8) predictor
    dot_predict<<<(int)((EL + 7) / 8), 256, 0, stream>>>(
        xu, xi, label_src, label_dst, out, EL);
}